// GATConv_72705206387211
// MI455X (gfx1250) — compile-verified
//
#include <hip/hip_runtime.h>

typedef float v2f __attribute__((ext_vector_type(2)));
typedef float v8f __attribute__((ext_vector_type(8)));

#define D 64

// ---- ordered-uint encoding for float atomicMax (handles negatives) ----
__device__ __forceinline__ unsigned int enc_f32(float f) {
  unsigned int u = __float_as_uint(f);
  return (u & 0x80000000u) ? ~u : (u | 0x80000000u);
}
__device__ __forceinline__ float dec_f32(unsigned int e) {
  unsigned int u = (e & 0x80000000u) ? (e ^ 0x80000000u) : ~e;
  return __uint_as_float(u);
}

// ---- pass 0: zero out / s, init m to encoded(-inf) ----
__global__ void __launch_bounds__(256) gat_init(float* __restrict__ out,
                                                float* __restrict__ s,
                                                unsigned int* __restrict__ mmax,
                                                int N) {
  int i = blockIdx.x * blockDim.x + threadIdx.x;
  if (i < N * D) out[i] = 0.0f;
  if (i < N) { s[i] = 0.0f; mmax[i] = 0x007FFFFFu; /* enc(-inf) */ }
}

// ---- pass 1: xl = x @ W + b via V_WMMA_F32_16X16X4_F32 ----
// 256 threads = 8 waves; each wave computes a 16x64 strip of xl.
__global__ void __launch_bounds__(256) gat_linear(const float* __restrict__ x,
                                                  const float* __restrict__ W,
                                                  const float* __restrict__ bias,
                                                  float* __restrict__ xl,
                                                  int N) {
  __shared__ float Ws[D * D];
  __shared__ float bs[D];
  int tid = threadIdx.x;
  for (int i = tid * 4; i < D * D; i += 256 * 4)
    *(float4*)&Ws[i] = *(const float4*)&W[i];
  if (tid < D) bs[tid] = bias[tid];
  __syncthreads();

  int wave = tid >> 5;
  int lane = tid & 31;
  int tile = blockIdx.x * 8 + wave;       // 16-row tile index
  int row0 = tile * 16;
  if (row0 >= N) return;                  // wave-uniform exit (EXEC stays full)

  int m  = lane & 15;                     // M (A) / N (B,C) index
  int kh = lane >> 4;                     // 0: K=0,1   1: K=2,3
  int rowA = row0 + m; if (rowA >= N) rowA = N - 1;
  const float* xr = x + (size_t)rowA * D;

  v8f acc[4] = {};
  for (int kk = 0; kk < D / 4; ++kk) {
    int kb = kk * 4 + kh * 2;
    v2f a;
    a.x = xr[kb + 0];
    a.y = xr[kb + 1];
#pragma unroll
    for (int nt = 0; nt < 4; ++nt) {
      v2f bb;
      bb.x = Ws[(kb + 0) * D + nt * 16 + m];
      bb.y = Ws[(kb + 1) * D + nt * 16 + m];
      acc[nt] = __builtin_amdgcn_wmma_f32_16x16x4_f32(
          /*neg_a=*/false, a, /*neg_b=*/false, bb,
          /*c_mod=*/(short)0, acc[nt], /*reuse_a=*/false, /*reuse_b=*/false);
    }
  }

  // C/D layout: VGPR v -> row v (lanes 0-15) / v+8 (lanes 16-31), col = lane&15
#pragma unroll
  for (int nt = 0; nt < 4; ++nt) {
#pragma unroll
    for (int v = 0; v < 8; ++v) {
      int r = row0 + v + 8 * kh;
      if (r < N)
        xl[(size_t)r * D + nt * 16 + m] = acc[nt][v] + bs[nt * 16 + m];
    }
  }
}

// ---- pass 2: alpha_raw[e] = leaky_relu(dot(xl[row], xl[col])); segment max ----
// 8 lanes per edge, float4 loads, shfl_xor reduction.
__global__ void __launch_bounds__(256) gat_score(const long long* __restrict__ rows,
                                                 const long long* __restrict__ cols,
                                                 const float* __restrict__ xl,
                                                 float* __restrict__ alpha,
                                                 unsigned int* __restrict__ mmax,
                                                 long long E) {
  long long g = (long long)blockIdx.x * blockDim.x + threadIdx.x;
  long long e = g >> 3;
  int t = (int)(g & 7);
  if (e >= E) return;
  long long r = rows[e], c = cols[e];
  const float4* xi = (const float4*)(xl + (size_t)r * D);
  const float4* xj = (const float4*)(xl + (size_t)c * D);
  float4 a0 = xi[t],     b0 = xj[t];
  float4 a1 = xi[t + 8], b1 = xj[t + 8];
  float p = a0.x * b0.x + a0.y * b0.y + a0.z * b0.z + a0.w * b0.w
          + a1.x * b1.x + a1.y * b1.y + a1.z * b1.z + a1.w * b1.w;
#pragma unroll
  for (int off = 1; off < 8; off <<= 1) p += __shfl_xor(p, off, 8);
  if (t == 0) {
    float a = (p >= 0.0f) ? p : 0.2f * p;   // leaky relu, slope 0.2
    alpha[e] = a;
    atomicMax(&mmax[r], enc_f32(a));
  }
}

// ---- pass 3: e = exp(alpha - m[row]); s[row] += e (in-place over alpha) ----
__global__ void __launch_bounds__(256) gat_exp(const long long* __restrict__ rows,
                                               float* __restrict__ alpha,
                                               const unsigned int* __restrict__ mmax,
                                               float* __restrict__ s,
                                               long long E) {
  long long e = (long long)blockIdx.x * blockDim.x + threadIdx.x;
  if (e >= E) return;
  long long r = rows[e];
  float mx = dec_f32(mmax[r]);
  if (!isfinite(mx)) mx = 0.0f;            // mirror reference neutralization
  float ev = expf(alpha[e] - mx);
  alpha[e] = ev;
  atomicAdd(&s[r], ev);
}

// ---- pass 4: out[row] += xl[col] * (e / (s[row]+1e-16)) ----
// one wave per edge; each lane handles dims lane and lane+32.
__global__ void __launch_bounds__(256) gat_aggregate(const long long* __restrict__ rows,
                                                     const long long* __restrict__ cols,
                                                     const float* __restrict__ xl,
                                                     const float* __restrict__ ealpha,
                                                     const float* __restrict__ s,
                                                     float* __restrict__ out,
                                                     long long E) {
  long long g = (long long)blockIdx.x * blockDim.x + threadIdx.x;
  long long e = g >> 5;
  int lane = (int)(g & 31);
  if (e >= E) return;
  long long r = rows[e], c = cols[e];
  float coeff = ealpha[e] / (s[r] + 1e-16f);
  const float* src = xl + (size_t)c * D;
  float* dst = out + (size_t)r * D;
  atomicAdd(&dst[lane],      src[lane]      * coeff);
  atomicAdd(&dst[lane + 32], src[lane + 32] * coeff);
}

extern "C" void kernel_launch(void* const* d_in, const int* in_sizes, int n_in,
                              void* d_out, int out_size, void* d_ws, size_t ws_size,
                              hipStream_t stream) {
  const float*      x  = (const float*)d_in[0];
  const long long*  ei = (const long long*)d_in[1];   // int64 [2, E]
  const float*      W  = (const float*)d_in[2];
  const float*      b  = (const float*)d_in[3];
  float*            out = (float*)d_out;

  int       N = in_sizes[0] / D;
  long long E = (long long)in_sizes[1] / 2;
  const long long* rows = ei;
  const long long* cols = ei + E;

  // workspace layout
  float*        xl    = (float*)d_ws;                   // N*D
  float*        alpha = xl + (size_t)N * D;             // E
  unsigned int* mmax  = (unsigned int*)(alpha + E);     // N
  float*        s     = (float*)(mmax + N);             // N

  {
    int total = N * D;
    gat_init<<<(total + 255) / 256, 256, 0, stream>>>(out, s, mmax, N);
  }
  {
    int tiles  = (N + 15) / 16;
    int blocks = (tiles + 7) / 8;
    gat_linear<<<blocks, 256, 0, stream>>>(x, W, b, xl, N);
  }
  {
    long long th = E * 8;
    gat_score<<<(int)((th + 255) / 256), 256, 0, stream>>>(rows, cols, xl, alpha, mmax, E);
  }
  {
    gat_exp<<<(int)((E + 255) / 256), 256, 0, stream>>>(rows, alpha, mmax, s, E);
  }
  {
    long long th = E * 32;
    gat_aggregate<<<(int)((th + 255) / 256), 256, 0, stream>>>(rows, cols, xl, alpha, s, out, E);
  }
}